// NeuralTuringMachine_3152505996096
// MI455X (gfx1250) — compile-verified
//
#include <hip/hip_runtime.h>
#include <math.h>

// Problem dims (compile-time constants from the reference)
#define Bsz 2048
#define INs 256
#define Cs  256
#define Ms  1024
#define Ds  32
#define OUTs 256
#define KCI (INs + Ds)   // 288, inner dim for [x | read_data]

typedef __attribute__((ext_vector_type(2))) float v2f;
typedef __attribute__((ext_vector_type(4))) float v4f;
typedef __attribute__((ext_vector_type(8))) float v8f;

static __device__ __forceinline__ v8f wmma_k4(v2f a, v2f b, v8f c) {
  // V_WMMA_F32_16X16X4_F32 : D = A(16x4) * B(4x16) + C(16x16), full f32
  return __builtin_amdgcn_wmma_f32_16x16x4_f32(false, a, false, b, (short)0, c,
                                               false, false);
}

static __device__ __forceinline__ float sigmoidf_(float x) {
  return 1.0f / (1.0f + expf(-x));
}

// ---------------------------------------------------------------------------
// K1: r = softmax(b_rattn) @ memory   (h==0 makes w_r batch-independent)
// ---------------------------------------------------------------------------
__global__ void read_vec_kernel(const float* __restrict__ b_rattn,
                                const float* __restrict__ memory,
                                float* __restrict__ r) {
  __shared__ float red[Ms];
  __shared__ float w[Ms];
  const int t = threadIdx.x;               // 1024 threads
  float v = b_rattn[t];
  red[t] = v;
  __syncthreads();
  for (int s = Ms / 2; s > 0; s >>= 1) {
    if (t < s) red[t] = fmaxf(red[t], red[t + s]);
    __syncthreads();
  }
  const float mx = red[0];
  __syncthreads();
  const float e = expf(v - mx);
  red[t] = e;
  __syncthreads();
  for (int s = Ms / 2; s > 0; s >>= 1) {
    if (t < s) red[t] = red[t] + red[t + s];
    __syncthreads();
  }
  const float inv_sum = 1.0f / red[0];
  w[t] = e * inv_sum;
  __syncthreads();
  if (t < Ds) {
    float acc = 0.0f;
    for (int m = 0; m < Ms; ++m) acc += w[m] * memory[m * Ds + t];
    r[t] = acc;
  }
}

// ---------------------------------------------------------------------------
// K2: gates = [x|r] @ W_ih.T + (b_ih+b_hh); fused LSTM activations -> h_new
//     (f-gate skipped: c_prev == 0). One wave = one 16(batch) x 16(chan) tile,
//     three accumulators (i, g, o gates).
// ---------------------------------------------------------------------------
__global__ void gates_hnew_kernel(const float* __restrict__ x,
                                  const float* __restrict__ r,
                                  const float* __restrict__ W_ih,
                                  const float* __restrict__ b_ih,
                                  const float* __restrict__ b_hh,
                                  float* __restrict__ h_new) {
  const int lane = threadIdx.x & 31;
  const int row0 = blockIdx.x * 16;           // batch tile
  const int c0 = blockIdx.y * 16;             // channel tile (0..255)
  const int mn = lane & 15;                   // M for A-frag, N for B-frag
  const int khalf = (lane >> 4) << 1;         // 0 or 2
  v8f acc_i = {}, acc_g = {}, acc_o = {};
  const float* xrow = x + (row0 + mn) * INs;
  const float* Wi = W_ih + (c0 + mn) * KCI;           // i-gate rows
  const float* Wg = W_ih + (512 + c0 + mn) * KCI;     // g-gate rows
  const float* Wo = W_ih + (768 + c0 + mn) * KCI;     // o-gate rows
  for (int it = 0; it < KCI / 4; ++it) {
    const int kb = it * 4 + khalf;            // even, so kb and kb+1 same side
    v2f a;
    if (kb < INs) { a.x = xrow[kb]; a.y = xrow[kb + 1]; }
    else          { a.x = r[kb - INs]; a.y = r[kb + 1 - INs]; }
    v2f bi; bi.x = Wi[kb]; bi.y = Wi[kb + 1];
    v2f bg; bg.x = Wg[kb]; bg.y = Wg[kb + 1];
    v2f bo; bo.x = Wo[kb]; bo.y = Wo[kb + 1];
    acc_i = wmma_k4(a, bi, acc_i);
    acc_g = wmma_k4(a, bg, acc_g);
    acc_o = wmma_k4(a, bo, acc_o);
  }
  const int col = c0 + mn;
  const float bias_i = b_ih[col] + b_hh[col];
  const float bias_g = b_ih[512 + col] + b_hh[512 + col];
  const float bias_o = b_ih[768 + col] + b_hh[768 + col];
  const int mbase = (lane >> 4) << 3;
#pragma unroll
  for (int v = 0; v < 8; ++v) {
    const int brow = row0 + mbase + v;
    const float ig = sigmoidf_(acc_i[v] + bias_i);
    const float gg = tanhf(acc_g[v] + bias_g);
    const float og = sigmoidf_(acc_o[v] + bias_o);
    h_new[brow * Cs + col] = og * tanhf(ig * gg);   // c_new = i*g (c_prev==0)
  }
}

// ---------------------------------------------------------------------------
// K3: logits = h_new @ W_wattn.T + b_wattn   [2048 x 1024], K=256
// ---------------------------------------------------------------------------
__global__ void wattn_logits_kernel(const float* __restrict__ h_new,
                                    const float* __restrict__ W_wattn,
                                    const float* __restrict__ b_wattn,
                                    float* __restrict__ logits) {
  const int lane = threadIdx.x & 31;
  const int row0 = blockIdx.x * 16;           // batch tile
  const int m0 = blockIdx.y * 16;             // memory-slot tile
  const int mn = lane & 15;
  const int khalf = (lane >> 4) << 1;
  v8f acc = {};
  const float* hrow = h_new + (row0 + mn) * Cs;
  const float* Wm = W_wattn + (m0 + mn) * Cs;
  for (int it = 0; it < Cs / 4; ++it) {
    const int kb = it * 4 + khalf;
    v2f a; a.x = hrow[kb]; a.y = hrow[kb + 1];
    v2f b; b.x = Wm[kb];  b.y = Wm[kb + 1];
    acc = wmma_k4(a, b, acc);
  }
  const float bias = b_wattn[m0 + mn];
  const int mbase = (lane >> 4) << 3;
#pragma unroll
  for (int v = 0; v < 8; ++v) {
    const int brow = row0 + mbase + v;
    logits[brow * Ms + m0 + mn] = acc[v] + bias;
  }
}

// ---------------------------------------------------------------------------
// K4: erase_v = sigmoid(h_new @ W_er.T + b_er), add_v = tanh(h_new @ W_ad.T + b_ad)
// ---------------------------------------------------------------------------
__global__ void eradd_kernel(const float* __restrict__ h_new,
                             const float* __restrict__ W_er,
                             const float* __restrict__ b_er,
                             const float* __restrict__ W_ad,
                             const float* __restrict__ b_ad,
                             float* __restrict__ erase_v,
                             float* __restrict__ add_v) {
  const int lane = threadIdx.x & 31;
  const int row0 = blockIdx.x * 16;
  const int c0 = blockIdx.y * 16;             // 0 or 16 (D=32)
  const int mn = lane & 15;
  const int khalf = (lane >> 4) << 1;
  v8f acc_e = {}, acc_a = {};
  const float* hrow = h_new + (row0 + mn) * Cs;
  const float* We = W_er + (c0 + mn) * Cs;
  const float* Wa = W_ad + (c0 + mn) * Cs;
  for (int it = 0; it < Cs / 4; ++it) {
    const int kb = it * 4 + khalf;
    v2f a;  a.x = hrow[kb]; a.y = hrow[kb + 1];
    v2f be; be.x = We[kb];  be.y = We[kb + 1];
    v2f ba; ba.x = Wa[kb];  ba.y = Wa[kb + 1];
    acc_e = wmma_k4(a, be, acc_e);
    acc_a = wmma_k4(a, ba, acc_a);
  }
  const int col = c0 + mn;
  const float bias_e = b_er[col];
  const float bias_a = b_ad[col];
  const int mbase = (lane >> 4) << 3;
#pragma unroll
  for (int v = 0; v < 8; ++v) {
    const int brow = row0 + mbase + v;
    erase_v[brow * Ds + col] = sigmoidf_(acc_e[v] + bias_e);
    add_v[brow * Ds + col] = tanhf(acc_a[v] + bias_a);
  }
}

// ---------------------------------------------------------------------------
// K5: output = [h_new | r] @ W_out.T + b_out   [2048 x 256], K=288
// ---------------------------------------------------------------------------
__global__ void outproj_kernel(const float* __restrict__ h_new,
                               const float* __restrict__ r,
                               const float* __restrict__ W_out,
                               const float* __restrict__ b_out,
                               float* __restrict__ out) {
  const int lane = threadIdx.x & 31;
  const int row0 = blockIdx.x * 16;
  const int c0 = blockIdx.y * 16;
  const int mn = lane & 15;
  const int khalf = (lane >> 4) << 1;
  v8f acc = {};
  const float* hrow = h_new + (row0 + mn) * Cs;
  const float* Wo = W_out + (c0 + mn) * KCI;
  for (int it = 0; it < KCI / 4; ++it) {
    const int kb = it * 4 + khalf;
    v2f a;
    if (kb < Cs) { a.x = hrow[kb]; a.y = hrow[kb + 1]; }
    else         { a.x = r[kb - Cs]; a.y = r[kb + 1 - Cs]; }
    v2f b; b.x = Wo[kb]; b.y = Wo[kb + 1];
    acc = wmma_k4(a, b, acc);
  }
  const float bias = b_out[c0 + mn];
  const int mbase = (lane >> 4) << 3;
#pragma unroll
  for (int v = 0; v < 8; ++v) {
    const int brow = row0 + mbase + v;
    out[brow * OUTs + c0 + mn] = acc[v] + bias;
  }
}

// ---------------------------------------------------------------------------
// K6: in-place row softmax over M=1024 (one block per batch row)
// ---------------------------------------------------------------------------
__global__ void softmax_rows_kernel(float* __restrict__ logits) {
  __shared__ float red[256];
  const int t = threadIdx.x;
  float* row = logits + (size_t)blockIdx.x * Ms;
  float v0 = row[t], v1 = row[t + 256], v2 = row[t + 512], v3 = row[t + 768];
  float mx = fmaxf(fmaxf(v0, v1), fmaxf(v2, v3));
  red[t] = mx;
  __syncthreads();
  for (int s = 128; s > 0; s >>= 1) {
    if (t < s) red[t] = fmaxf(red[t], red[t + s]);
    __syncthreads();
  }
  mx = red[0];
  __syncthreads();
  v0 = expf(v0 - mx); v1 = expf(v1 - mx); v2 = expf(v2 - mx); v3 = expf(v3 - mx);
  red[t] = v0 + v1 + v2 + v3;
  __syncthreads();
  for (int s = 128; s > 0; s >>= 1) {
    if (t < s) red[t] = red[t] + red[t + s];
    __syncthreads();
  }
  const float inv = 1.0f / red[0];
  row[t] = v0 * inv; row[t + 256] = v1 * inv;
  row[t + 512] = v2 * inv; row[t + 768] = v3 * inv;
}

// ---------------------------------------------------------------------------
// K7: write_data[b,m,d] = memory[m,d]*(1 - w_w[b,m]*erase_v[b,d]) + w_w[b,m]*add_v[b,d]
// 268 MB of stores -> the roofline limiter. One float4 per thread, perfectly
// coalesced, non-temporal stores (write-once data).
// ---------------------------------------------------------------------------
__global__ void write_data_kernel(const v4f* __restrict__ memory4,
                                  const float* __restrict__ w_w,
                                  const v4f* __restrict__ erase4,
                                  const v4f* __restrict__ add4,
                                  v4f* __restrict__ out4) {
  const unsigned idx = blockIdx.x * 256u + threadIdx.x;  // [0, 2048*1024*8)
  const int d4 = idx & 7;
  const int m = (idx >> 3) & (Ms - 1);
  const int b = idx >> 13;
  const float ww = w_w[b * Ms + m];
  const v4f mm = memory4[m * 8 + d4];
  const v4f ev = erase4[b * 8 + d4];
  const v4f av = add4[b * 8 + d4];
  v4f o;
  o.x = mm.x * (1.0f - ww * ev.x) + ww * av.x;
  o.y = mm.y * (1.0f - ww * ev.y) + ww * av.y;
  o.z = mm.z * (1.0f - ww * ev.z) + ww * av.z;
  o.w = mm.w * (1.0f - ww * ev.w) + ww * av.w;
  __builtin_nontemporal_store(o, &out4[idx]);
}

// ---------------------------------------------------------------------------
extern "C" void kernel_launch(void* const* d_in, const int* in_sizes, int n_in,
                              void* d_out, int out_size, void* d_ws, size_t ws_size,
                              hipStream_t stream) {
  (void)in_sizes; (void)n_in; (void)out_size; (void)ws_size;
  const float* x       = (const float*)d_in[0];
  const float* memory  = (const float*)d_in[1];
  // d_in[2] = W_rattn : dead (h == 0)
  const float* b_rattn = (const float*)d_in[3];
  const float* W_ih    = (const float*)d_in[4];
  // d_in[5] = W_hh : dead (h == 0)
  const float* b_ih    = (const float*)d_in[6];
  const float* b_hh    = (const float*)d_in[7];
  const float* W_wattn = (const float*)d_in[8];
  const float* b_wattn = (const float*)d_in[9];
  const float* W_er    = (const float*)d_in[10];
  const float* b_er    = (const float*)d_in[11];
  const float* W_ad    = (const float*)d_in[12];
  const float* b_ad    = (const float*)d_in[13];
  const float* W_out   = (const float*)d_in[14];
  const float* b_out   = (const float*)d_in[15];

  float* out = (float*)d_out;                      // [2048*256] then [2048*1024*32]
  float* ws = (float*)d_ws;
  float* r       = ws;                             // 64 (32 used, padded)
  float* h_new   = ws + 64;                        // 2048*256
  float* erase_v = h_new + Bsz * Cs;               // 2048*32
  float* add_v   = erase_v + Bsz * Ds;             // 2048*32
  float* logits  = add_v + Bsz * Ds;               // 2048*1024 (becomes w_w)

  read_vec_kernel<<<1, Ms, 0, stream>>>(b_rattn, memory, r);
  gates_hnew_kernel<<<dim3(Bsz / 16, Cs / 16), 32, 0, stream>>>(
      x, r, W_ih, b_ih, b_hh, h_new);
  wattn_logits_kernel<<<dim3(Bsz / 16, Ms / 16), 32, 0, stream>>>(
      h_new, W_wattn, b_wattn, logits);
  eradd_kernel<<<dim3(Bsz / 16, Ds / 16), 32, 0, stream>>>(
      h_new, W_er, b_er, W_ad, b_ad, erase_v, add_v);
  outproj_kernel<<<dim3(Bsz / 16, OUTs / 16), 32, 0, stream>>>(
      h_new, r, W_out, b_out, out);
  softmax_rows_kernel<<<Bsz, 256, 0, stream>>>(logits);
  write_data_kernel<<<(Bsz * Ms * (Ds / 4)) / 256, 256, 0, stream>>>(
      (const v4f*)memory, logits, (const v4f*)erase_v, (const v4f*)add_v,
      (v4f*)(out + Bsz * OUTs));
}